// GRU_74964359184998
// MI455X (gfx1250) — compile-verified
//
#include <hip/hip_runtime.h>
#include <hip/hip_bf16.h>
#include <math.h>

// ---------------------------------------------------------------------------
// Problem constants (from reference)
// ---------------------------------------------------------------------------
#define T_LEN 128
#define BATCH 64
#define EMB   512
#define HID   512
#define VOCAB 10000
#define NLAY  2

typedef float v2f __attribute__((ext_vector_type(2)));
typedef float v4f __attribute__((ext_vector_type(4)));
typedef float v8f __attribute__((ext_vector_type(8)));

// ---------------------------------------------------------------------------
// fp32 WMMA GEMM:  C = A(MxK) * B (+ bias broadcast over rows)
//   TRANS_B == false : B is KxN row-major
//   TRANS_B == true  : B is NxK row-major (compute A @ B^T)
// Block tile TBM x TBN, K staged via LDS in TBK chunks, 256 threads = 8
// wave32; each wave computes a TM x TN grid of 16x16 tiles with independent
// accumulators (ILP = TM*TN) via v_wmma_f32_16x16x4_f32. Global->LDS staging
// is software pipelined through registers. Requires K % TBK == 0 (K=512).
// ---------------------------------------------------------------------------
template <int TBM, int TBN, int TBK, int TM, int TN, bool TRANS_B, bool HAS_BIAS>
__global__ __launch_bounds__(256) void gemm_f32_wmma(
    const float* __restrict__ A, const float* __restrict__ B,
    const float* __restrict__ bias, float* __restrict__ C,
    int M, int N, int K)
{
    constexpr int PADA  = TBK + 1;
    constexpr int PADB  = TBN + 1;
    constexpr int WROWS = TBM / (16 * TM);          // waves down M
    constexpr int WCOLS = TBN / (16 * TN);          // waves across N
    static_assert(WROWS * WCOLS == 8, "need 8 waves");
    constexpr int CA = (TBM * TBK) / (256 * 4);     // float4 chunks per thread (A)
    constexpr int CB = (TBK * TBN) / (256 * 4);     // float4 chunks per thread (B)
    static_assert(CA >= 1 && CB >= 1, "tile too small");

    __shared__ float sA[TBM * PADA];                // [m][k]
    __shared__ float sB[TBK * PADB];                // [k][n]

    const int tid   = threadIdx.x;
    const int lane  = tid & 31;
    const int wave  = tid >> 5;
    const int lhalf = lane & 15;                    // 0..15
    const int khi   = lane >> 4;                    // 0 or 1

    const int m0 = blockIdx.y * TBM;
    const int n0 = blockIdx.x * TBN;

    const int wm = wave % WROWS;
    const int wn = wave / WROWS;

    v8f acc[TM][TN];
    #pragma unroll
    for (int tm = 0; tm < TM; ++tm)
        #pragma unroll
        for (int tn = 0; tn < TN; ++tn)
            acc[tm][tn] = (v8f){0.f, 0.f, 0.f, 0.f, 0.f, 0.f, 0.f, 0.f};

    v4f ra[CA], rb[CB];

    auto loadA = [&](int k0) {
        #pragma unroll
        for (int c = 0; c < CA; ++c) {
            const int id  = tid + c * 256;
            const int row = id / (TBK / 4);
            const int c4  = (id % (TBK / 4)) * 4;
            ra[c] = (v4f){0.f, 0.f, 0.f, 0.f};
            if (m0 + row < M)
                ra[c] = *(const v4f*)(A + (size_t)(m0 + row) * K + k0 + c4);
        }
    };
    auto loadB = [&](int k0) {
        if (TRANS_B) {
            #pragma unroll
            for (int c = 0; c < CB; ++c) {
                const int id = tid + c * 256;
                const int n  = id / (TBK / 4);
                const int ka = (id % (TBK / 4)) * 4;
                rb[c] = (v4f){0.f, 0.f, 0.f, 0.f};
                if (n0 + n < N)
                    rb[c] = *(const v4f*)(B + (size_t)(n0 + n) * K + k0 + ka);
            }
        } else {
            #pragma unroll
            for (int c = 0; c < CB; ++c) {
                const int id = tid + c * 256;
                const int k  = id / (TBN / 4);
                const int na = (id % (TBN / 4)) * 4;
                const float* src = B + (size_t)(k0 + k) * N + n0;
                v4f v = (v4f){0.f, 0.f, 0.f, 0.f};
                if (n0 + na + 3 < N) {
                    v = *(const v4f*)(src + na);
                } else {
                    #pragma unroll
                    for (int i = 0; i < 4; ++i)
                        if (n0 + na + i < N) v[i] = src[na + i];
                }
                rb[c] = v;
            }
        }
    };
    auto stageA = [&]() {
        #pragma unroll
        for (int c = 0; c < CA; ++c) {
            const int id  = tid + c * 256;
            const int row = id / (TBK / 4);
            const int c4  = (id % (TBK / 4)) * 4;
            float* d = sA + row * PADA + c4;
            d[0] = ra[c][0]; d[1] = ra[c][1]; d[2] = ra[c][2]; d[3] = ra[c][3];
        }
    };
    auto stageB = [&]() {
        if (TRANS_B) {
            #pragma unroll
            for (int c = 0; c < CB; ++c) {
                const int id = tid + c * 256;
                const int n  = id / (TBK / 4);
                const int ka = (id % (TBK / 4)) * 4;
                #pragma unroll
                for (int i = 0; i < 4; ++i)
                    sB[(ka + i) * PADB + n] = rb[c][i];
            }
        } else {
            #pragma unroll
            for (int c = 0; c < CB; ++c) {
                const int id = tid + c * 256;
                const int k  = id / (TBN / 4);
                const int na = (id % (TBN / 4)) * 4;
                float* d = sB + k * PADB + na;
                d[0] = rb[c][0]; d[1] = rb[c][1]; d[2] = rb[c][2]; d[3] = rb[c][3];
            }
        }
    };

    loadA(0);
    loadB(0);

    for (int k0 = 0; k0 < K; k0 += TBK) {
        stageA();
        stageB();
        __syncthreads();

        const int kn = k0 + TBK;
        if (kn < K) {                 // prefetch next stage while WMMAs run
            loadA(kn);
            loadB(kn);
        }

        #pragma unroll
        for (int kk = 0; kk < TBK; kk += 4) {
            const int ka = kk + (khi << 1);       // K pair for this half-wave
            v2f af[TM], bf[TN];
            #pragma unroll
            for (int tm = 0; tm < TM; ++tm) {
                const int row = (wm * TM + tm) * 16 + lhalf;
                af[tm].x = sA[row * PADA + ka];
                af[tm].y = sA[row * PADA + ka + 1];
            }
            #pragma unroll
            for (int tn = 0; tn < TN; ++tn) {
                const int cl = (wn * TN + tn) * 16 + lhalf;
                bf[tn].x = sB[ka * PADB + cl];
                bf[tn].y = sB[(ka + 1) * PADB + cl];
            }
            #pragma unroll
            for (int tm = 0; tm < TM; ++tm)
                #pragma unroll
                for (int tn = 0; tn < TN; ++tn)
                    acc[tm][tn] = __builtin_amdgcn_wmma_f32_16x16x4_f32(
                        false, af[tm], false, bf[tn], (short)0,
                        acc[tm][tn], false, false);
        }
        __syncthreads();
    }

    // ---- epilogue: C layout = VGPR r -> M = r + 8*khi, N = lhalf ----
    #pragma unroll
    for (int tm = 0; tm < TM; ++tm) {
        #pragma unroll
        for (int tn = 0; tn < TN; ++tn) {
            const int col = n0 + (wn * TN + tn) * 16 + lhalf;
            if (col < N) {
                const float bv = HAS_BIAS ? bias[col] : 0.0f;
                const int rowbase = m0 + (wm * TM + tm) * 16 + khi * 8;
                #pragma unroll
                for (int r = 0; r < 8; ++r) {
                    const int row = rowbase + r;
                    if (row < M)
                        C[(size_t)row * N + col] = acc[tm][tn][r] + bv;
                }
            }
        }
    }
}

// ---------------------------------------------------------------------------
// Embedding gather: one block per (t,b) row, 128 threads x float4 = 512 f32
// ---------------------------------------------------------------------------
__global__ __launch_bounds__(128) void embed_kernel(
    const int* __restrict__ tok, const float* __restrict__ table,
    float* __restrict__ dst)
{
    const int row = blockIdx.x;
    const int t   = tok[row];
    const v4f* src = (const v4f*)(table + (size_t)t * EMB);
    v4f* d = (v4f*)(dst + (size_t)row * EMB);
    d[threadIdx.x] = src[threadIdx.x];
}

// ---------------------------------------------------------------------------
// GRU gate elementwise: r = sig(Wrx+Urh), z = sig(Wzx+Uzh); emit r*h and z
// ---------------------------------------------------------------------------
__global__ __launch_bounds__(256) void gru_gates_kernel(
    const float* __restrict__ gx, const float* __restrict__ gh,
    const float* __restrict__ h, float* __restrict__ rh,
    float* __restrict__ zb)
{
    const int idx = blockIdx.x * 256 + threadIdx.x;   // 0 .. B*H-1
    const int b = idx >> 9;
    const int j = idx & (HID - 1);
    const float wrx = gx[b * (3 * HID) + j];
    const float wzx = gx[b * (3 * HID) + HID + j];
    const float urh = gh[b * (2 * HID) + j];
    const float uzh = gh[b * (2 * HID) + HID + j];
    const float r = 1.0f / (1.0f + expf(-(wrx + urh)));
    const float z = 1.0f / (1.0f + expf(-(wzx + uzh)));
    rh[idx] = r * h[idx];
    zb[idx] = z;
}

// ---------------------------------------------------------------------------
// GRU state update: h = (1-z)*h + z*tanh(Whx + htp)
// ---------------------------------------------------------------------------
__global__ __launch_bounds__(256) void gru_update_kernel(
    const float* __restrict__ gx, const float* __restrict__ htp,
    const float* __restrict__ zb, float* __restrict__ h)
{
    const int idx = blockIdx.x * 256 + threadIdx.x;
    const int b = idx >> 9;
    const int j = idx & (HID - 1);
    const float whx = gx[b * (3 * HID) + 2 * HID + j];
    const float ht  = tanhf(whx + htp[idx]);
    const float z   = zb[idx];
    h[idx] = (1.0f - z) * h[idx] + z * ht;
}

// ---------------------------------------------------------------------------
// Host-side launch helpers
//   big  : 128x128 block tile, 4x2 tiles/wave (large GEMMs: ILP 8,
//          1.5 fragment dwords per wmma)
//   small: 32x64 block tile, 1x1 tile/wave (latency-bound recurrent GEMMs,
//          maximizes block-level N-parallelism)
// ---------------------------------------------------------------------------
static inline void gemmN_big(const float* A, const float* B, const float* bias,
                             float* C, int M, int N, int K, hipStream_t s)
{
    dim3 g((N + 127) / 128, (M + 127) / 128);
    if (bias) gemm_f32_wmma<128, 128, 32, 4, 2, false, true><<<g, 256, 0, s>>>(A, B, bias, C, M, N, K);
    else      gemm_f32_wmma<128, 128, 32, 4, 2, false, false><<<g, 256, 0, s>>>(A, B, nullptr, C, M, N, K);
}
static inline void gemmT_big(const float* A, const float* B, const float* bias,
                             float* C, int M, int N, int K, hipStream_t s)
{
    dim3 g((N + 127) / 128, (M + 127) / 128);
    if (bias) gemm_f32_wmma<128, 128, 32, 4, 2, true, true><<<g, 256, 0, s>>>(A, B, bias, C, M, N, K);
    else      gemm_f32_wmma<128, 128, 32, 4, 2, true, false><<<g, 256, 0, s>>>(A, B, nullptr, C, M, N, K);
}
static inline void gemmN_sm(const float* A, const float* B, const float* bias,
                            float* C, int M, int N, int K, hipStream_t s)
{
    dim3 g((N + 63) / 64, (M + 31) / 32);
    if (bias) gemm_f32_wmma<32, 64, 32, 1, 1, false, true><<<g, 256, 0, s>>>(A, B, bias, C, M, N, K);
    else      gemm_f32_wmma<32, 64, 32, 1, 1, false, false><<<g, 256, 0, s>>>(A, B, nullptr, C, M, N, K);
}
static inline void gemmT_sm(const float* A, const float* B, const float* bias,
                            float* C, int M, int N, int K, hipStream_t s)
{
    dim3 g((N + 63) / 64, (M + 31) / 32);
    if (bias) gemm_f32_wmma<32, 64, 32, 1, 1, true, true><<<g, 256, 0, s>>>(A, B, bias, C, M, N, K);
    else      gemm_f32_wmma<32, 64, 32, 1, 1, true, false><<<g, 256, 0, s>>>(A, B, nullptr, C, M, N, K);
}

extern "C" void kernel_launch(void* const* d_in, const int* in_sizes, int n_in,
                              void* d_out, int out_size, void* d_ws, size_t ws_size,
                              hipStream_t stream)
{
    (void)in_sizes; (void)n_in; (void)out_size;

    const int*   inputs    = (const int*)d_in[0];    // (T, B)
    const float* hidden0   = (const float*)d_in[1];  // (L, B, H)
    const float* emb_table = (const float*)d_in[2];  // (V, E)
    const float* W_x       = (const float*)d_in[3];  // (L, H, 3H)
    const float* U_h       = (const float*)d_in[4];  // (L, H, 2H)
    const float* U_ht      = (const float*)d_in[5];  // (L, H, H)
    const float* b_rzh     = (const float*)d_in[6];  // (L, 3H)
    const float* fc_W      = (const float*)d_in[7];  // (L, H, H)
    const float* fc_b      = (const float*)d_in[8];  // (L, H)
    const float* dec_W     = (const float*)d_in[9];  // (V, H)
    const float* dec_b     = (const float*)d_in[10]; // (V)
    float* out = (float*)d_out;

    const size_t ROWS = (size_t)T_LEN * BATCH;       // 8192

    // ---- workspace carve-up (floats) ----
    float* ws = (float*)d_ws;
    size_t off = 0;
    float* embs = ws + off; off += ROWS * EMB;            // (T*B, E)
    float* Ybuf = ws + off; off += ROWS * HID;            // decoder input
    float* hbuf = ws + off; off += (size_t)NLAY * BATCH * HID;
    float* gh   = ws + off; off += (size_t)BATCH * 2 * HID;
    float* gxs  = ws + off; off += (size_t)BATCH * 3 * HID;
    float* rh   = ws + off; off += (size_t)BATCH * HID;
    float* zb   = ws + off; off += (size_t)BATCH * HID;
    float* htp  = ws + off; off += (size_t)BATCH * HID;
    float* inp1 = ws + off; off += (size_t)BATCH * HID;
    // optional big buffer: all layer-0 gates_x precomputed in one GEMM
    const bool pre = ws_size >= (off + ROWS * (size_t)(3 * HID)) * sizeof(float);
    float* Gx0 = ws + off;

    // layer strides
    const size_t WXs = (size_t)HID * 3 * HID;
    const size_t UHs = (size_t)HID * 2 * HID;
    const size_t UTs = (size_t)HID * HID;
    const size_t FCs = (size_t)HID * HID;

    // ---- h <- initial hidden ----
    hipMemcpyAsync(hbuf, hidden0, (size_t)NLAY * BATCH * HID * sizeof(float),
                   hipMemcpyDeviceToDevice, stream);

    // ---- embeddings ----
    embed_kernel<<<(unsigned)ROWS, 128, 0, stream>>>(inputs, emb_table, embs);

    // ---- layer-0 gates_x for all timesteps in one big WMMA GEMM ----
    if (pre)
        gemmN_big(embs, W_x, b_rzh, Gx0, (int)ROWS, 3 * HID, EMB, stream);

    const dim3 ewg((BATCH * HID) / 256);
    float* h0 = hbuf;
    float* h1 = hbuf + (size_t)BATCH * HID;

    for (int t = 0; t < T_LEN; ++t) {
        const float* x0 = embs + (size_t)t * BATCH * EMB;
        const float* gx0_t;
        if (pre) {
            gx0_t = Gx0 + (size_t)t * BATCH * (3 * HID);
        } else {
            gemmN_sm(x0, W_x, b_rzh, gxs, BATCH, 3 * HID, EMB, stream);
            gx0_t = gxs;
        }
        // ---- layer 0 ----
        gemmN_sm(h0, U_h, nullptr, gh, BATCH, 2 * HID, HID, stream);
        gru_gates_kernel<<<ewg, 256, 0, stream>>>(gx0_t, gh, h0, rh, zb);
        gemmN_sm(rh, U_ht, nullptr, htp, BATCH, HID, HID, stream);
        gru_update_kernel<<<ewg, 256, 0, stream>>>(gx0_t, htp, zb, h0);
        gemmT_sm(h0, fc_W, fc_b, inp1, BATCH, HID, HID, stream);

        // ---- layer 1 ----
        gemmN_sm(inp1, W_x + WXs, b_rzh + 3 * HID, gxs, BATCH, 3 * HID, HID, stream);
        gemmN_sm(h1, U_h + UHs, nullptr, gh, BATCH, 2 * HID, HID, stream);
        gru_gates_kernel<<<ewg, 256, 0, stream>>>(gxs, gh, h1, rh, zb);
        gemmN_sm(rh, U_ht + UTs, nullptr, htp, BATCH, HID, HID, stream);
        gru_update_kernel<<<ewg, 256, 0, stream>>>(gxs, htp, zb, h1);
        gemmT_sm(h1, fc_W + FCs, fc_b + HID, Ybuf + (size_t)t * BATCH * HID,
                 BATCH, HID, HID, stream);
    }

    // ---- decoder: logits = Y @ dec_W^T + dec_b  (8192 x 10000, K=512) ----
    gemmT_big(Ybuf, dec_W, dec_b, out, (int)ROWS, VOCAB, HID, stream);

    // ---- append h_final (L, B, H) after logits ----
    hipMemcpyAsync(out + ROWS * (size_t)VOCAB, hbuf,
                   (size_t)NLAY * BATCH * HID * sizeof(float),
                   hipMemcpyDeviceToDevice, stream);
}